// KD_Loss_9380208574774
// MI455X (gfx1250) — compile-verified
//
#include <hip/hip_runtime.h>

// Problem constants from the reference: (L, B, N, C) = (4, 16, 4096, 512).
// Only token j = N-1 contributes. Output: scalar f32.
#define L_DIM 4
#define B_DIM 16
#define N_DIM 4096
#define C_DIM 512
#define NROWS (L_DIM * B_DIM)       // 64 softmax rows
#define CH_PER_LANE (C_DIM / 32)    // 16 channels per lane (wave32)

typedef __attribute__((ext_vector_type(2))) float v2f;
typedef __attribute__((ext_vector_type(8))) float v8f;

// ---------------------------------------------------------------------------
// Kernel 1: one wave32 per (l,b) row. Each lane owns 16 contiguous channels,
// loaded as float4s (global_load_b128). All reductions are in-wave shuffle
// ladders (no LDS, no barriers). Writes one f32 partial per row to d_ws.
// sym-KL partial per row = 0.5 * sum_c (p1 - p2) * (lp1 - lp2).
// ---------------------------------------------------------------------------
__global__ __launch_bounds__(256) void kd_rows_kernel(
    const float* __restrict__ g1, const float* __restrict__ g2,
    float* __restrict__ partial) {
  const int wave = (int)(blockIdx.x * (blockDim.x >> 5) + (threadIdx.x >> 5));
  const int lane = (int)(threadIdx.x & 31);

  // row = l*B + b; element offset of last token's channel block for this lane
  const size_t base = (size_t)wave * (size_t)N_DIM * (size_t)C_DIM +
                      (size_t)(N_DIM - 1) * (size_t)C_DIM +
                      (size_t)lane * CH_PER_LANE;

  float x1[CH_PER_LANE], x2[CH_PER_LANE];
  const float4* p1 = (const float4*)(g1 + base);
  const float4* p2 = (const float4*)(g2 + base);
#pragma unroll
  for (int i = 0; i < CH_PER_LANE / 4; ++i) {
    float4 a = p1[i];
    float4 b = p2[i];
    x1[4 * i + 0] = a.x * 0.5f; x1[4 * i + 1] = a.y * 0.5f;
    x1[4 * i + 2] = a.z * 0.5f; x1[4 * i + 3] = a.w * 0.5f;
    x2[4 * i + 0] = b.x * 0.5f; x2[4 * i + 1] = b.y * 0.5f;
    x2[4 * i + 2] = b.z * 0.5f; x2[4 * i + 3] = b.w * 0.5f;
  }

  // Row max for both matrices (lane-local then wave shuffle ladder).
  float m1 = x1[0], m2 = x2[0];
#pragma unroll
  for (int i = 1; i < CH_PER_LANE; ++i) {
    m1 = fmaxf(m1, x1[i]);
    m2 = fmaxf(m2, x2[i]);
  }
#pragma unroll
  for (int mask = 16; mask >= 1; mask >>= 1) {
    m1 = fmaxf(m1, __shfl_xor(m1, mask, 32));
    m2 = fmaxf(m2, __shfl_xor(m2, mask, 32));
  }

  // Row sum of exp(x - max).
  float s1 = 0.0f, s2 = 0.0f;
#pragma unroll
  for (int i = 0; i < CH_PER_LANE; ++i) {
    s1 += __expf(x1[i] - m1);
    s2 += __expf(x2[i] - m2);
  }
#pragma unroll
  for (int mask = 16; mask >= 1; mask >>= 1) {
    s1 += __shfl_xor(s1, mask, 32);
    s2 += __shfl_xor(s2, mask, 32);
  }

  const float lse1 = m1 + __logf(s1);
  const float lse2 = m2 + __logf(s2);

  // Symmetric KL contribution: (p1 - p2) * (lp1 - lp2), summed over channels.
  float acc = 0.0f;
#pragma unroll
  for (int i = 0; i < CH_PER_LANE; ++i) {
    const float lp1 = x1[i] - lse1;
    const float lp2 = x2[i] - lse2;
    acc += (__expf(lp1) - __expf(lp2)) * (lp1 - lp2);
  }
#pragma unroll
  for (int mask = 16; mask >= 1; mask >>= 1) {
    acc += __shfl_xor(acc, mask, 32);
  }

  if (lane == 0) {
    partial[wave] = 0.5f * acc;
  }
}

// ---------------------------------------------------------------------------
// Kernel 2: exact FP32 reduction of the 64 row partials via
// V_WMMA_F32_16X16X4_F32. B (4x16 f32) holds exactly 64 values (2 VGPRs/lane,
// wave32); A = ones(16x4) gives D[m][n] = sum_k B[k][n]. Summing all of B is
// permutation invariant, so the precise (k,n)<->(lane,vgpr) mapping is
// irrelevant. d[0] holds two full D rows (M=0 in lanes 0-15, M=8 in lanes
// 16-31), each row equal to the column sums, so a 32-lane shuffle sum of d[0]
// yields 2 * total. Single full wave -> EXEC is all ones as WMMA requires.
// ---------------------------------------------------------------------------
__global__ __launch_bounds__(32) void kd_final_kernel(
    const float* __restrict__ partial, float* __restrict__ out) {
  const int lane = (int)threadIdx.x;

  v2f a;
  a[0] = 1.0f;
  a[1] = 1.0f;
  v2f b;
  b[0] = partial[lane];
  b[1] = partial[lane + 32];
  v8f c = {};

  // 8 args: (neg_a, A, neg_b, B, c_mod, C, reuse_a, reuse_b)
  v8f d = __builtin_amdgcn_wmma_f32_16x16x4_f32(
      /*neg_a=*/false, a, /*neg_b=*/false, b,
      /*c_mod=*/(short)0, c, /*reuse_a=*/false, /*reuse_b=*/false);

  float t = d[0];
#pragma unroll
  for (int mask = 16; mask >= 1; mask >>= 1) {
    t += __shfl_xor(t, mask, 32);
  }
  // t == 2 * sum(partials); output is mean over L of the per-l sums.
  if (lane == 0) {
    out[0] = t * (0.5f / (float)L_DIM);
  }
}

extern "C" void kernel_launch(void* const* d_in, const int* in_sizes, int n_in,
                              void* d_out, int out_size, void* d_ws, size_t ws_size,
                              hipStream_t stream) {
  (void)in_sizes; (void)n_in; (void)out_size; (void)ws_size;
  const float* g1 = (const float*)d_in[0];
  const float* g2 = (const float*)d_in[1];
  float* out = (float*)d_out;
  float* partial = (float*)d_ws;  // 64 floats of scratch

  // 64 rows, one wave32 per row: 8 blocks x 256 threads (8 waves each).
  kd_rows_kernel<<<NROWS / 8, 256, 0, stream>>>(g1, g2, partial);
  // Exact WMMA-based reduction of the 64 partials -> scalar.
  kd_final_kernel<<<1, 32, 0, stream>>>(partial, out);
}